// CPABTransform2D_BIS_23038204576190
// MI455X (gfx1250) — compile-verified
//
#include <hip/hip_runtime.h>
#include <cmath>

// CPAB 2D coupled-flow kernel for gfx1250 (MI455X).
//  - MLPs (1->10->10->10->9) run on the matrix pipe via V_WMMA_F32_16X16X4_F32,
//    computed transposed (A = W^T per-lane gathered, B = activations of 16 points),
//    two 16-point tiles per wave32, bias folded into the accumulator init.
//  - Layer-to-layer C-layout -> B-layout conversion via per-wave LDS staging
//    (row stride 18 floats -> conflict-free store/load bank groups). No
//    ds_bpermute: LDS load/store is EXEC-per-lane and immune to the
//    predication hazard the compiler created around cross-lane reads.
//  - All weight gathers are branchless (clamped address + select) so no
//    exec-mask diamonds are generated.
//  - CPA flow integration is scalar per lane (10 unrolled cell-hop steps,
//    v_log_f32 / v_exp_f32), cell coefficients staged in LDS for dynamic index.
//  - Null-space basis of the CPA constraint matrix is built host-side
//    (deterministic Gram-Schmidt; reference uses a LAPACK SVD basis of the
//    same space) and passed by value as a kernarg struct.

#define NCELLS 10
#define DTH 9

typedef __attribute__((ext_vector_type(2))) float v2f;
typedef __attribute__((ext_vector_type(8))) float v8f;

struct Basis { float bt[DTH][2 * NCELLS]; };  // bt[j][m]

// LDS staging geometry: 16 rows x 16 cols per tile, row stride 18 floats so
// that row r (lanes 0-15) and row r+8 (lanes 16-31) hit disjoint bank groups.
#define ROWS 18
#define TILESTRIDE (16 * ROWS)  // 288 floats per tile
#define WSTG (2 * TILESTRIDE)   // 576 floats per wave

__device__ __forceinline__ v8f wmma4(v2f a, v2f b, v8f c) {
  // D = A(16x4 f32) x B(4x16 f32) + C(16x16 f32)
  return __builtin_amdgcn_wmma_f32_16x16x4_f32(false, a, false, b, (short)0, c,
                                               false, false);
}

// Per-thread theta[9] = MLP(s) for this thread's own point, computed
// cooperatively across the wave with WMMA (two 16-point tiles).
__device__ __forceinline__ void mlp_theta(
    float s,
    const float* __restrict__ w0, const float* __restrict__ b0,
    const float* __restrict__ w1, const float* __restrict__ b1,
    const float* __restrict__ w2, const float* __restrict__ b2,
    const float* __restrict__ w3, const float* __restrict__ b3,
    float* __restrict__ stgW,   // per-wave staging, WSTG floats
    float* __restrict__ sbufW,  // per-wave input broadcast, 32 floats
    float theta[DTH]) {
  const int lane = threadIdx.x & 31;
  const int half = lane >> 4;
  const int col  = lane & 15;
  const int c10  = (col < 10) ? col : 0;
  const int c9   = (col < 9) ? col : 0;

  // Branchless per-lane weight gathers:
  // A-layout (16x4) element (M=col, K=4t+2*half+v) of W^T, i.e. W[k][col].
  float w0k[3][2], b0k[3][2], a1[3][2], a2[3][2], a3[3][2];
#pragma unroll
  for (int t = 0; t < 3; ++t) {
#pragma unroll
    for (int v = 0; v < 2; ++v) {
      const int k = 4 * t + 2 * half + v;
      const bool kin = (k < 10);
      const int kk = kin ? k : 0;
      const float l0 = w0[kk];
      const float lB = b0[kk];
      const float l1 = w1[kk * 10 + c10];
      const float l2 = w2[kk * 10 + c10];
      const float l3 = w3[kk * DTH + c9];
      w0k[t][v] = kin ? l0 : 0.0f;
      b0k[t][v] = kin ? lB : 0.0f;
      a1[t][v] = (kin && col < 10) ? l1 : 0.0f;
      a2[t][v] = (kin && col < 10) ? l2 : 0.0f;
      a3[t][v] = (kin && col < DTH) ? l3 : 0.0f;
    }
  }
  // Bias in C/D layout: VGPR r holds row m = r + 8*half, uniform over lanes.
  v8f cb1, cb2, cb3;
#pragma unroll
  for (int r = 0; r < 8; ++r) {
    const int m = r + 8 * half;
    const int m10 = (m < 10) ? m : 0;
    const int m9 = (m < DTH) ? m : 0;
    const float lb1 = b1[m10];
    const float lb2 = b2[m10];
    const float lb3 = b3[m9];
    cb1[r] = (m < 10) ? lb1 : 0.0f;
    cb2[r] = (m < 10) ? lb2 : 0.0f;
    cb3[r] = (m < DTH) ? lb3 : 0.0f;
  }

  // Broadcast per-point input scalars through LDS (wave-internal, in-order).
  sbufW[lane] = s;

#pragma unroll
  for (int tile = 0; tile < 2; ++tile) {
    float* stgT = stgW + tile * TILESTRIDE;
    const float sp = sbufW[tile * 16 + col];

    // Layer 1: H1 = relu(W1^T * H0 + b1), H0[k] = sp*w0[k]+b0[k]
    v8f acc = cb1;
#pragma unroll
    for (int t = 0; t < 3; ++t) {
      v2f A; A.x = a1[t][0]; A.y = a1[t][1];
      v2f Bv; Bv.x = fmaf(sp, w0k[t][0], b0k[t][0]);
              Bv.y = fmaf(sp, w0k[t][1], b0k[t][1]);
      acc = wmma4(A, Bv, acc);
    }
#pragma unroll
    for (int r = 0; r < 8; ++r) acc[r] = fmaxf(acc[r], 0.0f);

    // C-layout -> LDS (rows m>=10 are exactly zero by construction)
#pragma unroll
    for (int r = 0; r < 8; ++r) stgT[(r + 8 * half) * ROWS + col] = acc[r];

    // Layer 2: gather B-layout rows k = 4t+2*half+v from LDS
    v8f acc2 = cb2;
#pragma unroll
    for (int t = 0; t < 3; ++t) {
      v2f A; A.x = a2[t][0]; A.y = a2[t][1];
      v2f Bv; Bv.x = stgT[(4 * t + 2 * half + 0) * ROWS + col];
              Bv.y = stgT[(4 * t + 2 * half + 1) * ROWS + col];
      acc2 = wmma4(A, Bv, acc2);
    }
#pragma unroll
    for (int r = 0; r < 8; ++r) acc2[r] = fmaxf(acc2[r], 0.0f);

#pragma unroll
    for (int r = 0; r < 8; ++r) stgT[(r + 8 * half) * ROWS + col] = acc2[r];

    // Layer 3 (no relu): theta rows 0..8
    v8f acc3 = cb3;
#pragma unroll
    for (int t = 0; t < 3; ++t) {
      v2f A; A.x = a3[t][0]; A.y = a3[t][1];
      v2f Bv; Bv.x = stgT[(4 * t + 2 * half + 0) * ROWS + col];
              Bv.y = stgT[(4 * t + 2 * half + 1) * ROWS + col];
      acc3 = wmma4(A, Bv, acc3);
    }

    // theta rows to LDS for per-point extraction
#pragma unroll
    for (int r = 0; r < 8; ++r) stgT[(r + 8 * half) * ROWS + col] = acc3[r];
  }

  // Own point lives in tile = half, column = col.
#pragma unroll
  for (int j = 0; j < DTH; ++j)
    theta[j] = stgW[half * TILESTRIDE + j * ROWS + col];
}

// A[m] = sum_j theta[j] * BT[j][m]  (20 cell coefficients), staged to LDS.
__device__ __forceinline__ void compute_A(const float theta[DTH], const Basis& B,
                                          float* __restrict__ Ath) {
#pragma unroll
  for (int m = 0; m < 2 * NCELLS; ++m) {
    float acc = theta[0] * B.bt[0][m];
#pragma unroll
    for (int j = 1; j < DTH; ++j) acc = fmaf(theta[j], B.bt[j][m], acc);
    Ath[m] = acc;
  }
}

// Closed-form per-cell CPA flow integration with cell hopping (NCELLS steps).
__device__ __forceinline__ void cpab_flow(float x0, const float* __restrict__ Ath,
                                          float& phi_out, float& ld_out) {
  float phi = x0;
  float t = 1.0f;
  int c = (int)(phi * 10.0f);
  c = c < 0 ? 0 : (c > 9 ? 9 : c);
  float ld = 0.0f;
#pragma unroll
  for (int it = 0; it < NCELLS; ++it) {
    const float a = Ath[2 * c];
    const float b = Ath[2 * c + 1];
    const float v = fmaf(a, phi, b);
    const bool lin = fabsf(a) < 1e-10f;
    const float sa = lin ? 1.0f : a;
    const float xc = (v >= 0.0f ? (float)(c + 1) : (float)c) * 0.1f;
    const float bsa = b / sa;
    float den = phi + bsa;
    den = (fabsf(den) < 1e-30f) ? 1e-30f : den;
    const float ratio = (xc + bsa) / den;
    const float t_exp = __logf(fmaxf(ratio, 1e-30f)) / sa;
    const float sb = (fabsf(b) < 1e-30f) ? 1.0f : b;
    float thit = lin ? (xc - phi) / sb : t_exp;
    thit = (fabsf(v) < 1e-14f) ? __builtin_inff() : thit;
    const bool cross = thit < t;
    const float dt = cross ? thit : t;
    float phi_new = lin ? fmaf(b, dt, phi)
                        : fmaf(phi + bsa, __expf(a * dt), -bsa);
    phi_new = cross ? xc : phi_new;
    const bool active = t > 0.0f;
    phi = active ? phi_new : phi;
    ld += active ? a * dt : 0.0f;
    t = active ? t - dt : t;
    int cn = c + (v >= 0.0f ? 1 : -1);
    cn = cn < 0 ? 0 : (cn > 9 ? 9 : cn);
    c = (active && cross) ? cn : c;
  }
  phi_out = phi;
  ld_out = ld;
}

__global__ __launch_bounds__(256) void cpab2d_kernel(
    const float* __restrict__ x,
    const float* __restrict__ m1w0, const float* __restrict__ m1b0,
    const float* __restrict__ m1w1, const float* __restrict__ m1b1,
    const float* __restrict__ m1w2, const float* __restrict__ m1b2,
    const float* __restrict__ m1w3, const float* __restrict__ m1b3,
    const float* __restrict__ m2w0, const float* __restrict__ m2b0,
    const float* __restrict__ m2w1, const float* __restrict__ m2b1,
    const float* __restrict__ m2w2, const float* __restrict__ m2b2,
    const float* __restrict__ m2w3, const float* __restrict__ m2b3,
    float* __restrict__ out, int npts, Basis B) {
  __shared__ float ldsA[256 * 21];  // 20 coeffs/thread, stride 21 vs bank conflicts
  __shared__ float stg[8 * WSTG];   // per-wave WMMA layout staging
  __shared__ float sbuf[8 * 32];    // per-wave input broadcast

  const int tid = threadIdx.x;
  const int wave = tid >> 5;
  const int p = blockIdx.x * 256 + tid;

  float* stgW = &stg[wave * WSTG];
  float* sbufW = &sbuf[wave * 32];
  float* Ath = &ldsA[tid * 21];

  const float2 xv = reinterpret_cast<const float2*>(x)[p];
  const float EPSc = 1e-7f;
  const float x2 = fminf(fmaxf(xv.x, EPSc), 1.0f - EPSc);  // xs[:,1]
  const float x1 = fminf(fmaxf(xv.y, EPSc), 1.0f - EPSc);  // xs[:,0]

  float theta[DTH];

  // theta2 = MLP_m2(x1); z2, ld2 = flow(x2, theta2)
  mlp_theta(x1, m2w0, m2b0, m2w1, m2b1, m2w2, m2b2, m2w3, m2b3, stgW, sbufW,
            theta);
  compute_A(theta, B, Ath);
  float z2, ld2;
  cpab_flow(x2, Ath, z2, ld2);

  // theta1 = MLP_m1(z2); z1, ld1 = flow(x1, theta1)
  mlp_theta(z2, m1w0, m1b0, m1w1, m1b1, m1w2, m1b2, m1w3, m1b3, stgW, sbufW,
            theta);
  compute_A(theta, B, Ath);
  float z1, ld1;
  cpab_flow(x1, Ath, z1, ld1);

  // out: z (N,2) then log|dz/dx| (N,2), both column-swapped back: [f(x2), f(x1)]
  float2* oz = reinterpret_cast<float2*>(out);
  oz[p] = make_float2(z2, z1);
  oz[npts + p] = make_float2(ld2, ld1);
}

extern "C" void kernel_launch(void* const* d_in, const int* in_sizes, int n_in,
                              void* d_out, int out_size, void* d_ws, size_t ws_size,
                              hipStream_t stream) {
  (void)n_in; (void)out_size; (void)d_ws; (void)ws_size;

  // Deterministic null-space basis of the CPA constraint matrix L (11x20):
  // continuity at cell boundaries, v(0)=0, v(1)=0. Free params a_0..a_8,
  // then Gram-Schmidt. (Reference uses a LAPACK SVD basis of the same space.)
  Basis B;
  double V[DTH][2 * NCELLS];
  for (int e = 0; e < DTH; ++e) {
    double a[NCELLS] = {0.0}, b[NCELLS] = {0.0};
    a[e] = 1.0;
    for (int k = 1; k <= 8; ++k) b[k] = b[k - 1] + (k / 10.0) * (a[k - 1] - a[k]);
    a[9] = -10.0 * (b[8] + 0.9 * a[8]);
    b[9] = b[8] + 0.9 * (a[8] - a[9]);
    for (int i = 0; i < NCELLS; ++i) {
      V[e][2 * i] = a[i];
      V[e][2 * i + 1] = b[i];
    }
  }
  for (int e = 0; e < DTH; ++e) {
    for (int f = 0; f < e; ++f) {
      double d = 0.0;
      for (int m = 0; m < 20; ++m) d += V[e][m] * V[f][m];
      for (int m = 0; m < 20; ++m) V[e][m] -= d * V[f][m];
    }
    double n = 0.0;
    for (int m = 0; m < 20; ++m) n += V[e][m] * V[e][m];
    n = std::sqrt(n);
    for (int m = 0; m < 20; ++m) V[e][m] /= n;
  }
  for (int j = 0; j < DTH; ++j)
    for (int m = 0; m < 20; ++m) B.bt[j][m] = (float)V[j][m];

  const float* xin = (const float*)d_in[0];
  const int npts = in_sizes[0] / 2;  // 2097152
  dim3 grid((unsigned)(npts / 256)), block(256);
  cpab2d_kernel<<<grid, block, 0, stream>>>(
      xin,
      (const float*)d_in[1], (const float*)d_in[2], (const float*)d_in[3],
      (const float*)d_in[4], (const float*)d_in[5], (const float*)d_in[6],
      (const float*)d_in[7], (const float*)d_in[8],
      (const float*)d_in[9], (const float*)d_in[10], (const float*)d_in[11],
      (const float*)d_in[12], (const float*)d_in[13], (const float*)d_in[14],
      (const float*)d_in[15], (const float*)d_in[16],
      (float*)d_out, npts, B);
}